// MemoryEfficientGNN_16123307229577
// MI455X (gfx1250) — compile-verified
//
#include <hip/hip_runtime.h>
#include <math.h>

#define NNODES 100000
#define NEDGES 1600000
#define IN_DIM 128
#define POS_DIM 16
#define CONTENT_DIM 112
#define NH 4
#define HD 32
#define HD4 8
#define OUTCOLS 128  // NH*HD
#define BPAD 114     // padded LDS row stride (floats): even (8B-aligned b64) + bank-conflict-free

typedef __attribute__((ext_vector_type(2))) float v2f;
typedef __attribute__((ext_vector_type(8))) float v8f;

// -------- order-preserving float <-> uint encoding for atomicMax ------------
__device__ __forceinline__ unsigned enc_f32(float f) {
  unsigned b = __float_as_uint(f);
  return (b & 0x80000000u) ? ~b : (b | 0x80000000u);
}
__device__ __forceinline__ float dec_f32(unsigned u) {
  return __uint_as_float((u & 0x80000000u) ? (u & 0x7fffffffu) : ~u);
}

// ---------------------------------------------------------------------------
// Kernel 0: zero the accumulators (workspace/d_out are poisoned by harness)
// ---------------------------------------------------------------------------
__global__ void init_kernel(unsigned* __restrict__ emax_u,
                            float* __restrict__ esum,
                            float* __restrict__ out) {
  size_t total = (size_t)NNODES * OUTCOLS + 1;
  size_t stride = (size_t)gridDim.x * blockDim.x;
  for (size_t i = (size_t)blockIdx.x * blockDim.x + threadIdx.x; i < total; i += stride) {
    out[i] = 0.0f;
    if (i < (size_t)NNODES * NH) {
      esum[i] = 0.0f;
      emax_u[i] = 0u;  // encodes "below every real float"
    }
  }
}

// ---------------------------------------------------------------------------
// Kernel 1: hc = content @ Wc^T via V_WMMA_F32_16X16X4_F32
//   M = NNODES (100000, multiple of 32), K = 112, Nout = 128
//   Block: 256 threads = 8 waves; covers 32 rows x 128 cols.
//   Wave w: mtile = w/4 (16 rows), ntile = w%4 (32 cols -> two 16x16 tiles).
//   Wc staged in LDS in native row-major layout (padded stride 114 floats):
//   the K-adjacent pair a lane needs for the B operand is one ds_load_b64.
// ---------------------------------------------------------------------------
__global__ __launch_bounds__(256) void gemm_hc_kernel(
    const float* __restrict__ feat, const float* __restrict__ Wc,
    float* __restrict__ hc) {
  __shared__ __align__(16) float sB[OUTCOLS * BPAD];  // 128*114*4 = 57 KB

  const int tid = threadIdx.x;
  for (int idx = tid; idx < OUTCOLS * CONTENT_DIM; idx += 256) {
    int j = idx / CONTENT_DIM;
    int k = idx - j * CONTENT_DIM;
    sB[j * BPAD + k] = Wc[idx];  // contiguous, coalesced read of Wc
  }
  __syncthreads();

  const int wave = tid >> 5;
  const int lane = tid & 31;
  const int m0 = blockIdx.x * 32 + (wave >> 2) * 16;
  const int n0 = (wave & 3) * 32;

  const int lm = lane & 15;        // row (A) / col (B,C) within 16
  const int kh = (lane >> 4) * 2;  // k sub-pair: lanes 0-15 -> k0,k1; 16-31 -> k2,k3

  v8f c0 = {0.f, 0.f, 0.f, 0.f, 0.f, 0.f, 0.f, 0.f};
  v8f c1 = {0.f, 0.f, 0.f, 0.f, 0.f, 0.f, 0.f, 0.f};

  const float* arow = feat + (size_t)(m0 + lm) * IN_DIM;  // content = cols 0..111
  const float* brow0 = sB + (n0 + lm) * BPAD;
  const float* brow1 = sB + (n0 + 16 + lm) * BPAD;

  for (int k0 = 0; k0 < CONTENT_DIM; k0 += 4) {
    v2f a  = *(const v2f*)(arow + k0 + kh);   // global_load_b64 (8B aligned)
    v2f b0 = *(const v2f*)(brow0 + k0 + kh);  // ds_load_b64 (8B aligned)
    v2f b1 = *(const v2f*)(brow1 + k0 + kh);  // ds_load_b64 (8B aligned)
    c0 = __builtin_amdgcn_wmma_f32_16x16x4_f32(false, a, false, b0, (short)0, c0, false, false);
    c1 = __builtin_amdgcn_wmma_f32_16x16x4_f32(false, a, false, b1, (short)0, c1, false, false);
  }

  // C/D layout: VGPR v -> row = v + 8*(lane/16), col = lane%16
  const int row_half = (lane >> 4) * 8;
#pragma unroll
  for (int v = 0; v < 8; ++v) {
    int row = m0 + row_half + v;
    hc[(size_t)row * OUTCOLS + n0 + lm] = c0[v];
    hc[(size_t)row * OUTCOLS + n0 + 16 + lm] = c1[v];
  }
}

// ---------------------------------------------------------------------------
// Kernel 2: per-node attention logits folded into qsrc/qdst (N x H each)
// ---------------------------------------------------------------------------
__global__ __launch_bounds__(256) void node_scores_kernel(
    const float* __restrict__ feat, const float* __restrict__ hc,
    const float* __restrict__ Wp,
    const float* __restrict__ attn_src, const float* __restrict__ attn_dst,
    const float* __restrict__ pos_attn_src, const float* __restrict__ pos_attn_dst,
    const float* __restrict__ att_comb,
    float* __restrict__ qsrc, float* __restrict__ qdst) {
  int n = blockIdx.x * blockDim.x + threadIdx.x;
  if (n >= NNODES) return;

  const float* hrow = hc + (size_t)n * OUTCOLS;
  float posv[POS_DIM];
#pragma unroll
  for (int k = 0; k < POS_DIM; ++k)
    posv[k] = feat[(size_t)n * IN_DIM + CONTENT_DIM + k];

#pragma unroll
  for (int h = 0; h < NH; ++h) {
    float es = 0.f, ed = 0.f;
#pragma unroll
    for (int d = 0; d < HD; ++d) {
      float v = hrow[h * HD + d];
      es += v * attn_src[h * HD + d];
      ed += v * attn_dst[h * HD + d];
    }
    float ps = 0.f, pd = 0.f;
#pragma unroll
    for (int d4 = 0; d4 < HD4; ++d4) {
      float hp = 0.f;
#pragma unroll
      for (int k = 0; k < POS_DIM; ++k)
        hp += posv[k] * Wp[(h * HD4 + d4) * POS_DIM + k];
      ps += hp * pos_attn_src[h * HD4 + d4];
      pd += hp * pos_attn_dst[h * HD4 + d4];
    }
    float cc0 = att_comb[h * 2 + 0];
    float cc1 = att_comb[h * 2 + 1];
    qsrc[n * NH + h] = es * cc0 + ps * cc1;
    qdst[n * NH + h] = ed * cc0 + pd * cc1;
  }
}

// ---------------------------------------------------------------------------
// Kernel 3: per-edge logit -> leaky relu -> segment max (uint-encoded atomicMax)
// ---------------------------------------------------------------------------
__global__ __launch_bounds__(256) void edge_max_kernel(
    const int* __restrict__ src, const int* __restrict__ dst,
    const float* __restrict__ qsrc, const float* __restrict__ qdst,
    unsigned* __restrict__ emax_u) {
  int i = blockIdx.x * blockDim.x + threadIdx.x;
  if (i >= NEDGES) return;
  int s = src[i], t = dst[i];
#pragma unroll
  for (int h = 0; h < NH; ++h) {
    float e = qsrc[s * NH + h] + qdst[t * NH + h];
    e = (e > 0.f) ? e : 0.2f * e;
    atomicMax(&emax_u[t * NH + h], enc_f32(e));
  }
}

// ---------------------------------------------------------------------------
// Kernel 4: one wave32 per edge: ex = exp(e - emax[dst]);
//           esum[dst][h] += ex;  out[dst][h*32+lane] += ex * hc[src][h*32+lane]
// ---------------------------------------------------------------------------
__global__ __launch_bounds__(256) void edge_accum_kernel(
    const int* __restrict__ src, const int* __restrict__ dst,
    const float* __restrict__ qsrc, const float* __restrict__ qdst,
    const unsigned* __restrict__ emax_u, const float* __restrict__ hc,
    float* __restrict__ esum, float* __restrict__ out) {
  const int lane = threadIdx.x & 31;
  long long i = (long long)blockIdx.x * 8 + (threadIdx.x >> 5);
  if (i >= NEDGES) return;
  int s = src[i], t = dst[i];

  float ex[NH];
#pragma unroll
  for (int h = 0; h < NH; ++h) {
    float e = qsrc[s * NH + h] + qdst[t * NH + h];
    e = (e > 0.f) ? e : 0.2f * e;
    float m = dec_f32(emax_u[t * NH + h]);
    ex[h] = __expf(e - m);
  }
  if (lane == 0) {
#pragma unroll
    for (int h = 0; h < NH; ++h) atomicAdd(&esum[t * NH + h], ex[h]);
  }
#pragma unroll
  for (int h = 0; h < NH; ++h) {
    float val = ex[h] * hc[(size_t)s * OUTCOLS + h * HD + lane];
    atomicAdd(&out[(size_t)t * OUTCOLS + h * HD + lane], val);
  }
}

// ---------------------------------------------------------------------------
// Kernel 5: out = (esum>0 ? acc/esum : 0) + feat; diversity loss = 0
// ---------------------------------------------------------------------------
__global__ __launch_bounds__(256) void finalize_kernel(
    const float* __restrict__ feat, const float* __restrict__ esum,
    float* __restrict__ out) {
  size_t total = (size_t)NNODES * OUTCOLS;
  size_t i = (size_t)blockIdx.x * blockDim.x + threadIdx.x;
  if (i == 0) out[total] = 0.0f;  // diversity_loss
  if (i >= total) return;
  int n = (int)(i >> 7);
  int h = ((int)i & 127) >> 5;
  float s = esum[n * NH + h];
  float acc = out[i];
  float v = (s > 0.f) ? (acc / s) : 0.f;
  out[i] = v + feat[i];
}

// ---------------------------------------------------------------------------
extern "C" void kernel_launch(void* const* d_in, const int* in_sizes, int n_in,
                              void* d_out, int out_size, void* d_ws, size_t ws_size,
                              hipStream_t stream) {
  const float* feat         = (const float*)d_in[0];
  const int*   src          = (const int*)d_in[1];
  const int*   dst          = (const int*)d_in[2];
  const float* Wc           = (const float*)d_in[3];
  const float* Wp           = (const float*)d_in[4];
  const float* attn_src     = (const float*)d_in[5];
  const float* attn_dst     = (const float*)d_in[6];
  const float* pos_attn_src = (const float*)d_in[7];
  const float* pos_attn_dst = (const float*)d_in[8];
  const float* att_comb     = (const float*)d_in[9];
  float* out = (float*)d_out;

  // workspace layout
  float*    hc     = (float*)d_ws;                             // N*128
  float*    qsrc   = hc + (size_t)NNODES * OUTCOLS;            // N*4
  float*    qdst   = qsrc + (size_t)NNODES * NH;               // N*4
  unsigned* emax_u = (unsigned*)(qdst + (size_t)NNODES * NH);  // N*4
  float*    esum   = (float*)(emax_u + (size_t)NNODES * NH);   // N*4

  // 0) zero accumulators + d_out
  {
    size_t total = (size_t)NNODES * OUTCOLS + 1;
    int blocks = (int)((total + 255) / 256);
    init_kernel<<<blocks, 256, 0, stream>>>(emax_u, esum, out);
  }
  // 1) hc GEMM (WMMA f32)
  gemm_hc_kernel<<<NNODES / 32, 256, 0, stream>>>(feat, Wc, hc);
  // 2) per-node scores
  node_scores_kernel<<<(NNODES + 255) / 256, 256, 0, stream>>>(
      feat, hc, Wp, attn_src, attn_dst, pos_attn_src, pos_attn_dst, att_comb,
      qsrc, qdst);
  // 3) segment max
  edge_max_kernel<<<(NEDGES + 255) / 256, 256, 0, stream>>>(src, dst, qsrc, qdst, emax_u);
  // 4) exp + weighted scatter (one wave per edge)
  edge_accum_kernel<<<(NEDGES + 7) / 8, 256, 0, stream>>>(
      src, dst, qsrc, qdst, emax_u, hc, esum, out);
  // 5) normalize + residual
  finalize_kernel<<<((size_t)NNODES * OUTCOLS + 255) / 256, 256, 0, stream>>>(feat, esum, out);
}